// MultiHeadAttention_20813411516938
// MI455X (gfx1250) — compile-verified
//
#include <hip/hip_runtime.h>
#include <hip/hip_bf16.h>

typedef __attribute__((ext_vector_type(2))) float v2f;
typedef __attribute__((ext_vector_type(8))) float v8f;

#define SQ   2048      // sequence length
#define DM   1024      // embed dim
#define NH   16        // heads
#define DH   64        // head dim
#define BT   2         // batch
#define MROWS (BT*SQ)  // 4096 rows for dense GEMMs

__device__ __forceinline__ v8f wmma4(v2f a, v2f b, v8f c) {
  // D = A(16x4 f32) * B(4x16 f32) + C(16x16 f32)
  return __builtin_amdgcn_wmma_f32_16x16x4_f32(
      /*neg_a=*/false, a, /*neg_b=*/false, b,
      /*c_mod=*/(short)0, c, /*reuse_a=*/false, /*reuse_b=*/false);
}

// ---------------------------------------------------------------------------
// Dense GEMM + bias: C[4096x1024] = A[4096x1024] * W[1024x1024] + bias
// block = 128 threads (4 waves), tile 64(M) x 64(N), BK = 32.
// Double-buffered LDS, one barrier per K-stage; global loads for stage k+1
// issue while stage k computes. waves_per_eu(1) relaxes the VGPR budget so
// the staged tile registers don't spill to scratch.
// ---------------------------------------------------------------------------
__global__ __launch_bounds__(128)
__attribute__((amdgpu_waves_per_eu(1)))
void gemm_bias_kernel(
    const float* __restrict__ A, const float* __restrict__ W,
    const float* __restrict__ bias, float* __restrict__ C)
{
  __shared__ float As[2][64 * 36];   // [64][32] pad->36
  __shared__ v2f   Bs2[2][16 * 80];  // [16 kp][64 col] pad->80

  const int tid  = threadIdx.x;
  const int wave = tid >> 5;
  const int lane = tid & 31;
  const int laneRow = lane & 15;
  const int laneK   = lane >> 4;
  const int m0 = blockIdx.y * 64;
  const int n0 = blockIdx.x * 64;

  // per-thread staging coordinates (scalars, no local arrays)
  const int rA0 = tid >> 3,        cA0 = (tid & 7) * 4;        // A rows 0..15
  const int kpB = tid >> 4,        cgB = (tid & 15) * 4;       // first 256? tid<128 -> kp 0..7
  const int kpB1 = (tid + 128) >> 4, cgB1 = ((tid + 128) & 15) * 4; // kp 8..15

  const v8f vzero = {0.f,0.f,0.f,0.f,0.f,0.f,0.f,0.f};
  v8f acc[4] = {vzero, vzero, vzero, vzero};

  float4 ra0, ra1, ra2, ra3, rb0a, rb0b, rb1a, rb1b;
  // prefetch stage 0 into registers
  {
    const float* Abase = A + (size_t)(m0 + rA0) * DM + cA0;
    ra0 = *(const float4*)(Abase);
    ra1 = *(const float4*)(Abase + 16 * DM);
    ra2 = *(const float4*)(Abase + 32 * DM);
    ra3 = *(const float4*)(Abase + 48 * DM);
    const float* s0 = W + (size_t)(2 * kpB)  * DM + n0 + cgB;
    const float* s1 = W + (size_t)(2 * kpB1) * DM + n0 + cgB1;
    rb0a = *(const float4*)(s0); rb0b = *(const float4*)(s0 + DM);
    rb1a = *(const float4*)(s1); rb1b = *(const float4*)(s1 + DM);
  }

  int buf = 0;
  for (int k0 = 0; k0 < DM; k0 += 32, buf ^= 1) {
    // commit staged registers to LDS buffer buf
    float* Aw = &As[buf][rA0 * 36 + cA0];
    *(float4*)(Aw            ) = ra0;
    *(float4*)(Aw + 16 * 36  ) = ra1;
    *(float4*)(Aw + 32 * 36  ) = ra2;
    *(float4*)(Aw + 48 * 36  ) = ra3;
    {
      v2f* d0 = &Bs2[buf][kpB * 80 + cgB];
      v2f t0 = {rb0a.x, rb0b.x}; v2f t1 = {rb0a.y, rb0b.y};
      v2f t2 = {rb0a.z, rb0b.z}; v2f t3 = {rb0a.w, rb0b.w};
      d0[0] = t0; d0[1] = t1; d0[2] = t2; d0[3] = t3;
      v2f* d1 = &Bs2[buf][kpB1 * 80 + cgB1];
      v2f u0 = {rb1a.x, rb1b.x}; v2f u1 = {rb1a.y, rb1b.y};
      v2f u2 = {rb1a.z, rb1b.z}; v2f u3 = {rb1a.w, rb1b.w};
      d1[0] = u0; d1[1] = u1; d1[2] = u2; d1[3] = u3;
    }
    __syncthreads();

    // issue next stage's global loads (latency overlapped with WMMA below)
    if (k0 + 32 < DM) {
      const int kn = k0 + 32;
      const float* Abase = A + (size_t)(m0 + rA0) * DM + kn + cA0;
      ra0 = *(const float4*)(Abase);
      ra1 = *(const float4*)(Abase + 16 * DM);
      ra2 = *(const float4*)(Abase + 32 * DM);
      ra3 = *(const float4*)(Abase + 48 * DM);
      const float* s0 = W + (size_t)(kn + 2 * kpB)  * DM + n0 + cgB;
      const float* s1 = W + (size_t)(kn + 2 * kpB1) * DM + n0 + cgB1;
      rb0a = *(const float4*)(s0); rb0b = *(const float4*)(s0 + DM);
      rb1a = *(const float4*)(s1); rb1b = *(const float4*)(s1 + DM);
    }

    const float* Ap = &As[buf][(wave * 16 + laneRow) * 36];
    const v2f*   Bp = &Bs2[buf][0];
    #pragma unroll
    for (int k4 = 0; k4 < 32; k4 += 4) {
      int ka = k4 + 2 * laneK;
      v2f a = *(const v2f*)(Ap + ka);
      int kp = (k4 >> 1) + laneK;
      #pragma unroll
      for (int nb = 0; nb < 4; ++nb)
        acc[nb] = wmma4(a, Bp[kp * 80 + nb * 16 + laneRow], acc[nb]);
    }
  }

  const int rbase = m0 + wave * 16 + 8 * laneK;   // C layout: M = r + 8*(lane>=16)
  #pragma unroll
  for (int nb = 0; nb < 4; ++nb) {
    int col = n0 + nb * 16 + laneRow;
    float bv = bias[col];
    #pragma unroll
    for (int r = 0; r < 8; ++r)
      C[(size_t)(rbase + r) * DM + col] = acc[nb][r] + bv;
  }
}

// ---------------------------------------------------------------------------
// scores[bh][q][k] = (Q . K^T) * 0.125 + mask * -1e9    (raw, pre-softmax)
// grid (S/64, S/64, B*H); K staged transposed AND pair-interleaved:
// Ks2[kp][n] = (K[n][2kp], K[n][2kp+1])  -- contiguous in global memory.
// Single K-stage (K = DH = 64), no pipelining needed.
// ---------------------------------------------------------------------------
__global__ __launch_bounds__(128) void scores_kernel(
    const float* __restrict__ Q, const float* __restrict__ Kmat,
    const int* __restrict__ mask, float* __restrict__ attn)
{
  __shared__ float Qs[64 * 68];   // [row][k] pad 64->68
  __shared__ v2f   Ks2[32 * 80];  // [32 kp][64 n] pad->80

  const int bh = blockIdx.z;
  const int b  = bh >> 4;         // /NH
  const int h  = bh & 15;
  const float* Qb = Q    + (size_t)b * SQ * DM + h * DH;
  const float* Kb = Kmat + (size_t)b * SQ * DM + h * DH;
  float*       Cb = attn + (size_t)bh * SQ * SQ;

  const int tid = threadIdx.x;
  const int wave = tid >> 5, lane = tid & 31;
  const int laneRow = lane & 15, laneK = lane >> 4;
  const int m0 = blockIdx.y * 64, n0 = blockIdx.x * 64;

  #pragma unroll
  for (int i = 0; i < 8; ++i) {                   // Q tile 64x64
    int t = tid + i * 128;
    int r = t >> 4, c = (t & 15) * 4;
    float4 v = *(const float4*)(Qb + (size_t)(m0 + r) * DM + c);
    *(float4*)(&Qs[r * 68 + c]) = v;
  }
  #pragma unroll
  for (int i = 0; i < 8; ++i) {                   // K tile, transpose + interleave
    int t = tid + i * 128;
    int n = t >> 4;                               // key row 0..63
    int kq = (t & 15) * 4;                        // k offset 0..60
    float4 v = *(const float4*)(Kb + (size_t)(n0 + n) * DM + kq);
    v2f lo = {v.x, v.y};
    v2f hi = {v.z, v.w};
    Ks2[((kq >> 1) + 0) * 80 + n] = lo;
    Ks2[((kq >> 1) + 1) * 80 + n] = hi;
  }
  __syncthreads();

  const v8f vzero = {0.f,0.f,0.f,0.f,0.f,0.f,0.f,0.f};
  v8f acc[4] = {vzero, vzero, vzero, vzero};
  const float* Ap = &Qs[(wave * 16 + laneRow) * 68];
  #pragma unroll
  for (int k4 = 0; k4 < 64; k4 += 4) {
    int ka = k4 + 2 * laneK;
    v2f a = *(const v2f*)(Ap + ka);
    int kp = (k4 >> 1) + laneK;
    #pragma unroll
    for (int nb = 0; nb < 4; ++nb)
      acc[nb] = wmma4(a, Ks2[kp * 80 + nb * 16 + laneRow], acc[nb]);
  }

  const int* mrow = mask + (size_t)b * SQ;
  const int rbase = m0 + wave * 16 + 8 * laneK;
  #pragma unroll
  for (int nb = 0; nb < 4; ++nb) {
    int col = n0 + nb * 16 + laneRow;
    float mf = (float)mrow[col] * -1e9f;
    #pragma unroll
    for (int r = 0; r < 8; ++r)
      Cb[(size_t)(rbase + r) * SQ + col] = acc[nb][r] * 0.125f + mf;
  }
}

// ---------------------------------------------------------------------------
// In-place row softmax over the last axis (row length 2048), 1 block per row
// ---------------------------------------------------------------------------
__global__ __launch_bounds__(256) void softmax_kernel(float* __restrict__ attn)
{
  __shared__ float red[256];
  float* row = attn + (size_t)blockIdx.x * SQ;
  const int tid = threadIdx.x;

  float v[8];
  float m = -3.4e38f;
  #pragma unroll
  for (int i = 0; i < 8; ++i) { v[i] = row[tid + i * 256]; m = fmaxf(m, v[i]); }
  red[tid] = m; __syncthreads();
  for (int s = 128; s > 0; s >>= 1) {
    if (tid < s) red[tid] = fmaxf(red[tid], red[tid + s]);
    __syncthreads();
  }
  m = red[0]; __syncthreads();

  float sum = 0.f;
  #pragma unroll
  for (int i = 0; i < 8; ++i) { v[i] = __expf(v[i] - m); sum += v[i]; }
  red[tid] = sum; __syncthreads();
  for (int s = 128; s > 0; s >>= 1) {
    if (tid < s) red[tid] += red[tid + s];
    __syncthreads();
  }
  float inv = 1.0f / red[0];
  #pragma unroll
  for (int i = 0; i < 8; ++i) row[tid + i * 256] = v[i] * inv;
}

// ---------------------------------------------------------------------------
// ctx = attn * V, written directly into (B,S,D) "concat" layout
// grid (1, S/64, B*H); M=2048, N=64, K=2048, BK=32. Double-buffered like
// the dense GEMM (64 K-stages -> pipelining matters most here).
// ---------------------------------------------------------------------------
__global__ __launch_bounds__(128)
__attribute__((amdgpu_waves_per_eu(1)))
void ctx_kernel(
    const float* __restrict__ attn, const float* __restrict__ V,
    float* __restrict__ ctx)
{
  __shared__ float As[2][64 * 36];
  __shared__ v2f   Bs2[2][16 * 80];

  const int bh = blockIdx.z;
  const int b  = bh >> 4, h = bh & 15;
  const float* Ab = attn + (size_t)bh * SQ * SQ;          // row stride SQ
  const float* Vb = V    + (size_t)b * SQ * DM + h * DH;  // row stride DM
  float*       Cb = ctx  + (size_t)b * SQ * DM + h * DH;  // concat layout

  const int tid = threadIdx.x;
  const int wave = tid >> 5, lane = tid & 31;
  const int laneRow = lane & 15, laneK = lane >> 4;
  const int m0 = blockIdx.y * 64;

  const int rA0 = tid >> 3,        cA0 = (tid & 7) * 4;
  const int kpB = tid >> 4,        cgB = (tid & 15) * 4;
  const int kpB1 = (tid + 128) >> 4, cgB1 = ((tid + 128) & 15) * 4;

  const v8f vzero = {0.f,0.f,0.f,0.f,0.f,0.f,0.f,0.f};
  v8f acc[4] = {vzero, vzero, vzero, vzero};

  float4 ra0, ra1, ra2, ra3, rb0a, rb0b, rb1a, rb1b;
  {
    const float* Abase = Ab + (size_t)(m0 + rA0) * SQ + cA0;
    ra0 = *(const float4*)(Abase);
    ra1 = *(const float4*)(Abase + 16 * SQ);
    ra2 = *(const float4*)(Abase + 32 * SQ);
    ra3 = *(const float4*)(Abase + 48 * SQ);
    const float* s0 = Vb + (size_t)(2 * kpB)  * DM + cgB;
    const float* s1 = Vb + (size_t)(2 * kpB1) * DM + cgB1;
    rb0a = *(const float4*)(s0); rb0b = *(const float4*)(s0 + DM);
    rb1a = *(const float4*)(s1); rb1b = *(const float4*)(s1 + DM);
  }

  int buf = 0;
  for (int k0 = 0; k0 < SQ; k0 += 32, buf ^= 1) {
    float* Aw = &As[buf][rA0 * 36 + cA0];
    *(float4*)(Aw          ) = ra0;
    *(float4*)(Aw + 16 * 36) = ra1;
    *(float4*)(Aw + 32 * 36) = ra2;
    *(float4*)(Aw + 48 * 36) = ra3;
    {
      v2f* d0 = &Bs2[buf][kpB * 80 + cgB];
      v2f t0 = {rb0a.x, rb0b.x}; v2f t1 = {rb0a.y, rb0b.y};
      v2f t2 = {rb0a.z, rb0b.z}; v2f t3 = {rb0a.w, rb0b.w};
      d0[0] = t0; d0[1] = t1; d0[2] = t2; d0[3] = t3;
      v2f* d1 = &Bs2[buf][kpB1 * 80 + cgB1];
      v2f u0 = {rb1a.x, rb1b.x}; v2f u1 = {rb1a.y, rb1b.y};
      v2f u2 = {rb1a.z, rb1b.z}; v2f u3 = {rb1a.w, rb1b.w};
      d1[0] = u0; d1[1] = u1; d1[2] = u2; d1[3] = u3;
    }
    __syncthreads();

    if (k0 + 32 < SQ) {
      const int kn = k0 + 32;
      const float* Abase = Ab + (size_t)(m0 + rA0) * SQ + kn + cA0;
      ra0 = *(const float4*)(Abase);
      ra1 = *(const float4*)(Abase + 16 * SQ);
      ra2 = *(const float4*)(Abase + 32 * SQ);
      ra3 = *(const float4*)(Abase + 48 * SQ);
      const float* s0 = Vb + (size_t)(kn + 2 * kpB)  * DM + cgB;
      const float* s1 = Vb + (size_t)(kn + 2 * kpB1) * DM + cgB1;
      rb0a = *(const float4*)(s0); rb0b = *(const float4*)(s0 + DM);
      rb1a = *(const float4*)(s1); rb1b = *(const float4*)(s1 + DM);
    }

    const float* Ap = &As[buf][(wave * 16 + laneRow) * 36];
    const v2f*   Bp = &Bs2[buf][0];
    #pragma unroll
    for (int k4 = 0; k4 < 32; k4 += 4) {
      int ka = k4 + 2 * laneK;
      v2f a = *(const v2f*)(Ap + ka);
      int kp = (k4 >> 1) + laneK;
      #pragma unroll
      for (int nb = 0; nb < 4; ++nb)
        acc[nb] = wmma4(a, Bp[kp * 80 + nb * 16 + laneRow], acc[nb]);
    }
  }

  const int rbase = m0 + wave * 16 + 8 * laneK;
  #pragma unroll
  for (int nb = 0; nb < 4; ++nb) {
    int col = nb * 16 + laneRow;                  // 0..63 within head slice
    #pragma unroll
    for (int r = 0; r < 8; ++r)
      Cb[(size_t)(rbase + r) * DM + col] = acc[nb][r];
  }
}

// ---------------------------------------------------------------------------
extern "C" void kernel_launch(void* const* d_in, const int* in_sizes, int n_in,
                              void* d_out, int out_size, void* d_ws, size_t ws_size,
                              hipStream_t stream) {
  const float* x    = (const float*)d_in[0];
  const int*   mask = (const int*)  d_in[1];
  const float* Wq   = (const float*)d_in[2];
  const float* bq   = (const float*)d_in[3];
  const float* Wk   = (const float*)d_in[4];
  const float* bk   = (const float*)d_in[5];
  const float* Wv   = (const float*)d_in[6];
  const float* bv   = (const float*)d_in[7];
  const float* Wo   = (const float*)d_in[8];
  const float* bo   = (const float*)d_in[9];

  float* out  = (float*)d_out;
  float* attn = out + (size_t)BT * SQ * DM;       // second tuple output

  float* q   = (float*)d_ws;
  float* k   = q   + (size_t)BT * SQ * DM;
  float* v   = k   + (size_t)BT * SQ * DM;
  float* ctx = v   + (size_t)BT * SQ * DM;

  dim3 gDense(DM / 64, MROWS / 64, 1);            // (16, 64)
  gemm_bias_kernel<<<gDense, 128, 0, stream>>>(x, Wq, bq, q);
  gemm_bias_kernel<<<gDense, 128, 0, stream>>>(x, Wk, bk, k);
  gemm_bias_kernel<<<gDense, 128, 0, stream>>>(x, Wv, bv, v);

  scores_kernel<<<dim3(SQ / 64, SQ / 64, BT * NH), 128, 0, stream>>>(q, k, mask, attn);
  softmax_kernel<<<BT * NH * SQ, 256, 0, stream>>>(attn);
  ctx_kernel<<<dim3(1, SQ / 64, BT * NH), 128, 0, stream>>>(attn, v, ctx);

  gemm_bias_kernel<<<gDense, 128, 0, stream>>>(ctx, Wo, bo, out);
}